// ScreenTriangulation_80075370266852
// MI455X (gfx1250) — compile-verified
//
#include <hip/hip_runtime.h>
#include <hip/hip_bf16.h>
#include <stdint.h>

typedef float float4v __attribute__((ext_vector_type(4)));

#define THREADS 256
#define UBATCH  16   // b128 chunks per lane per phase -> 64KB LDS staging per block

// ---------------------------------------------------------------------------
// Bulk 256MB feature copy via CDNA5 async global<->LDS DMA path (ASYNCcnt).
// Phase structure: issue UBATCH async b128 loads into private LDS slots,
// drain ASYNCcnt, issue UBATCH async b128 stores, drain again before LDS
// reuse. Per-wave 8KB in flight per phase; grid supplies the concurrency.
// Single-pass stream => non-temporal hints on both sides.
// ---------------------------------------------------------------------------
__global__ __launch_bounds__(THREADS)
void feat_copy_async(const float4v* __restrict__ src,
                     float4v* __restrict__ dst, long long n4) {
    __shared__ float4v lds[THREADS * UBATCH];   // 64 KB staging
    const long long T   = (long long)gridDim.x * THREADS;
    const long long tid = (long long)blockIdx.x * THREADS + threadIdx.x;
    // Low 32 bits of the generic address of a __shared__ object == LDS offset.
    unsigned lds_base = (unsigned)(uintptr_t)(&lds[threadIdx.x]);

    for (long long base = tid; base < n4; base += T * UBATCH) {
#pragma unroll
        for (int j = 0; j < UBATCH; ++j) {
            long long idx = base + (long long)j * T;
            if (idx < n4) {
                unsigned laddr = lds_base + (unsigned)(j * THREADS * 16);
                const float4v* g = src + idx;
                asm volatile("global_load_async_to_lds_b128 %0, %1, off th:TH_LOAD_NT"
                             :: "v"(laddr), "v"(g) : "memory");
            }
        }
        asm volatile("s_wait_asynccnt 0x0" ::: "memory");
#pragma unroll
        for (int j = 0; j < UBATCH; ++j) {
            long long idx = base + (long long)j * T;
            if (idx < n4) {
                unsigned laddr = lds_base + (unsigned)(j * THREADS * 16);
                float4v* g = dst + idx;
                asm volatile("global_store_async_from_lds_b128 %0, %1, off th:TH_STORE_NT"
                             :: "v"(g), "v"(laddr) : "memory");
            }
        }
        asm volatile("s_wait_asynccnt 0x0" ::: "memory");
    }
}

// ---------------------------------------------------------------------------
// verts_3d = [2*sigmoid(logit)-1 (rows 0..3 pinned to boundary), ones column]
// ---------------------------------------------------------------------------
__global__ void verts_kernel(const float* __restrict__ logit,
                             const float* __restrict__ boundary,
                             float* __restrict__ out, int V) {
    int i = blockIdx.x * blockDim.x + threadIdx.x;
    if (i >= V) return;
    float x = logit[2 * i + 0];
    float y = logit[2 * i + 1];
    float a = 2.0f / (1.0f + expf(-x)) - 1.0f;
    float b = 2.0f / (1.0f + expf(-y)) - 1.0f;
    if (i < 4) {
        a = boundary[2 * i + 0];
        b = boundary[2 * i + 1];
    }
    out[3 * i + 0] = a;
    out[3 * i + 1] = b;
    out[3 * i + 2] = 1.0f;
}

// ---------------------------------------------------------------------------
// faces: int32 -> float32 cast-copy (d_out is a float buffer)
// ---------------------------------------------------------------------------
__global__ void faces_kernel(const int* __restrict__ faces,
                             float* __restrict__ out, int n) {
    int i = blockIdx.x * blockDim.x + threadIdx.x;
    if (i < n) out[i] = (float)faces[i];
}

// Fallback when the destination offset is not 16B-aligned (data-dependent F).
__global__ void feat_copy_plain(const float* __restrict__ src,
                                float* __restrict__ dst, long long n) {
    long long T   = (long long)gridDim.x * blockDim.x;
    long long tid = (long long)blockIdx.x * blockDim.x + threadIdx.x;
    for (long long i = tid; i < n; i += T) dst[i] = src[i];
}

extern "C" void kernel_launch(void* const* d_in, const int* in_sizes, int n_in,
                              void* d_out, int out_size, void* d_ws, size_t ws_size,
                              hipStream_t stream) {
    const float* logit    = (const float*)d_in[0];   // (V,2)
    const float* feats    = (const float*)d_in[1];   // (V,128)
    const float* boundary = (const float*)d_in[2];   // (4,2)
    const int*   faces    = (const int*)d_in[3];     // (F,3)

    const int       V     = in_sizes[0] / 2;
    const long long nfeat = (long long)in_sizes[1];
    const int       nf3   = in_sizes[3];

    float* out       = (float*)d_out;
    float* out_faces = out + (long long)3 * V;
    float* out_feat  = out_faces + nf3;

    verts_kernel<<<(V + THREADS - 1) / THREADS, THREADS, 0, stream>>>(
        logit, boundary, out, V);
    faces_kernel<<<(nf3 + THREADS - 1) / THREADS, THREADS, 0, stream>>>(
        faces, out_faces, nf3);

    const long long featOffElems = (long long)3 * V + nf3;
    if (((featOffElems & 3) == 0) && ((nfeat & 3) == 0)) {
        long long n4 = nfeat / 4;
        feat_copy_async<<<2048, THREADS, 0, stream>>>(
            (const float4v*)feats, (float4v*)out_feat, n4);
    } else {
        feat_copy_plain<<<4096, THREADS, 0, stream>>>(feats, out_feat, nfeat);
    }
}